// Cheb_Conv_28123445854491
// MI455X (gfx1250) — compile-verified
//
#include <hip/hip_runtime.h>

typedef __attribute__((ext_vector_type(2))) float v2f;
typedef __attribute__((ext_vector_type(8))) float v8f;

#define F_IN   64      // input feats
#define K_CHEB 4
#define F_CAT  (F_IN * K_CHEB)   // 256
#define F_OUT  256
#define MT     5       // M-tiles per wave (50000 = 625 * 80, exact)

// ---------------------------------------------------------------- degree
__global__ void k_degree(const int* __restrict__ dst, float* __restrict__ deg, int nE) {
    int e = blockIdx.x * blockDim.x + threadIdx.x;
    if (e < nE) unsafeAtomicAdd(&deg[dst[e]], 1.0f);
}

// deg -> rsqrt(max(deg,1)) in place
__global__ void k_dsqrt(float* __restrict__ deg, int n) {
    int i = blockIdx.x * blockDim.x + threadIdx.x;
    if (i < n) {
        float d = deg[i];
        d = d < 1.0f ? 1.0f : d;
        deg[i] = rsqrtf(d);
    }
}

// ---------------------------------------------------------------- X0 copy (signal -> Xt[:, 0:64])
__global__ void k_copy_x0(const float* __restrict__ sig, float* __restrict__ Xt, int n) {
    int t = blockIdx.x * blockDim.x + threadIdx.x;      // n*16 threads, one float4 each
    if (t < n * 16) {
        int i = t >> 4, c = t & 15;
        const float4 v = reinterpret_cast<const float4*>(sig)[(size_t)i * 16 + c];
        reinterpret_cast<float4*>(Xt)[(size_t)i * 64 + c] = v;  // row stride 256 floats
    }
}

// ---------------------------------------------------------------- edge scatter: agg[dst] += x[src]*dinv[src]
// x points at a 64-wide column block inside Xt (row stride 256 floats)
__global__ void k_scatter(const int* __restrict__ src, const int* __restrict__ dst,
                          const float* __restrict__ x, const float* __restrict__ dinv,
                          float* __restrict__ agg, int nE) {
    int t = blockIdx.x * blockDim.x + threadIdx.x;      // nE*16 threads, one float4 each
    if (t >= nE * 16) return;
    int e = t >> 4, c = t & 15;
    int s  = src[e];
    int d0 = dst[e];
    float ds = dinv[s];
    const float4 v = *reinterpret_cast<const float4*>(x + (size_t)s * F_CAT + c * 4);
    float* ap = agg + (size_t)d0 * F_IN + c * 4;
    unsafeAtomicAdd(ap + 0, v.x * ds);
    unsafeAtomicAdd(ap + 1, v.y * ds);
    unsafeAtomicAdd(ap + 2, v.z * ds);
    unsafeAtomicAdd(ap + 3, v.w * ds);
}

// ---------------------------------------------------------------- Chebyshev combine
// Xout = alpha * (agg * dinv) + beta * Xprev + gamma * Xpp
__global__ void k_combine(const float* __restrict__ agg, const float* __restrict__ dinv,
                          const float* __restrict__ lambda_max,
                          const float* __restrict__ Xprev, const float* __restrict__ Xpp,
                          float* __restrict__ Xout, int n, int first_step) {
    int t = blockIdx.x * blockDim.x + threadIdx.x;      // n*16 threads, one float4 each
    if (t >= n * 16) return;
    int i = t >> 4, c = t & 15;
    float rn = 2.0f / lambda_max[0];
    float alpha, beta, gamma;
    if (first_step) { alpha = -rn;        beta = rn - 1.0f;          gamma = 0.0f;  }
    else            { alpha = -2.0f * rn; beta = 2.0f * (rn - 1.0f); gamma = -1.0f; }
    float di = dinv[i];
    const float4 a   = *reinterpret_cast<const float4*>(agg   + (size_t)i * F_IN  + c * 4);
    const float4 xp  = *reinterpret_cast<const float4*>(Xprev + (size_t)i * F_CAT + c * 4);
    const float4 xpp = *reinterpret_cast<const float4*>(Xpp   + (size_t)i * F_CAT + c * 4);
    float4 o;
    o.x = alpha * (a.x * di) + beta * xp.x + gamma * xpp.x;
    o.y = alpha * (a.y * di) + beta * xp.y + gamma * xpp.y;
    o.z = alpha * (a.z * di) + beta * xp.z + gamma * xpp.z;
    o.w = alpha * (a.w * di) + beta * xp.w + gamma * xpp.w;
    *reinterpret_cast<float4*>(Xout + (size_t)i * F_CAT + c * 4) = o;
}

// ---------------------------------------------------------------- GEMM: out = relu(Xt[M,256] @ W[256,256] + b)
// block = 256 threads = 8 waves; each wave owns MT stacked 16x16 output tiles
// (one B fragment feeds MT WMMAs). grid = (M/(16*MT), 256/(8*16)).
__global__ __launch_bounds__(256) void k_gemm(const float* __restrict__ Xt,
                                              const float* __restrict__ W,
                                              const float* __restrict__ bias,
                                              float* __restrict__ out, int M) {
    const int wave   = threadIdx.x >> 5;
    const int lane   = threadIdx.x & 31;
    const int laneLo = lane & 15;
    const int hi     = lane >> 4;           // 0 or 1

    const int m0 = blockIdx.x * (16 * MT);
    const int n0 = blockIdx.y * 128 + wave * 16;

    // A fragment base per M-tile (clamped so tail tiles read in-bounds)
    const float* abase[MT];
#pragma unroll
    for (int i = 0; i < MT; ++i) {
        int mrow = m0 + i * 16 + laneLo;
        if (mrow >= M) mrow = M - 1;
        abase[i] = Xt + (size_t)mrow * F_CAT + 2 * hi;
    }
    // B fragment: lane holds W[k + 2*hi + {0,1}][n0 + laneLo]
    const float* bbase = W + (size_t)(2 * hi) * F_OUT + n0 + laneLo;

    v8f acc[MT];
#pragma unroll
    for (int i = 0; i < MT; ++i) acc[i] = (v8f){};

#pragma unroll 4
    for (int k = 0; k < F_CAT; k += 4) {
        v2f bb;
        bb.x = bbase[(size_t)(k + 0) * F_OUT];
        bb.y = bbase[(size_t)(k + 1) * F_OUT];
#pragma unroll
        for (int i = 0; i < MT; ++i) {
            v2f a = *reinterpret_cast<const v2f*>(abase[i] + k);
            acc[i] = __builtin_amdgcn_wmma_f32_16x16x4_f32(
                false, a, false, bb, (short)0, acc[i], false, false);
        }
    }

    const float bv = bias[n0 + laneLo];
    // D layout: VGPR r -> row (r + 8*hi), col laneLo
    if (m0 + 16 * MT <= M) {
        // full-tile fast path: unconditional store clause
#pragma unroll
        for (int i = 0; i < MT; ++i) {
            float* ob = out + (size_t)(m0 + i * 16 + 8 * hi) * F_OUT + n0 + laneLo;
#pragma unroll
            for (int r = 0; r < 8; ++r) {
                float v = acc[i][r] + bv;
                ob[(size_t)r * F_OUT] = v > 0.0f ? v : 0.0f;
            }
        }
    } else {
        // ragged tail (never taken for M=50000)
#pragma unroll
        for (int i = 0; i < MT; ++i) {
            float* ob = out + (size_t)(m0 + i * 16 + 8 * hi) * F_OUT + n0 + laneLo;
#pragma unroll
            for (int r = 0; r < 8; ++r) {
                int m = m0 + i * 16 + 8 * hi + r;
                if (m < M) {
                    float v = acc[i][r] + bv;
                    ob[(size_t)r * F_OUT] = v > 0.0f ? v : 0.0f;
                }
            }
        }
    }
}

// ----------------------------------------------------------------
static inline size_t align_up(size_t x, size_t a) { return (x + a - 1) & ~(a - 1); }

extern "C" void kernel_launch(void* const* d_in, const int* in_sizes, int n_in,
                              void* d_out, int out_size, void* d_ws, size_t ws_size,
                              hipStream_t stream) {
    const float* signal     = (const float*)d_in[0];
    const int*   src        = (const int*)d_in[1];
    const int*   dst        = (const int*)d_in[2];
    const float* lambda_max = (const float*)d_in[3];
    const float* W          = (const float*)d_in[4];
    const float* bias       = (const float*)d_in[5];
    float*       out        = (float*)d_out;

    const int nNodes = in_sizes[0] / F_IN;   // 50000
    const int nEdges = in_sizes[1];          // 800000

    // workspace carve-up
    char*  ws   = (char*)d_ws;
    size_t off  = 0;
    float* dinv = (float*)(ws + off); off = align_up(off + (size_t)nNodes * sizeof(float), 256);
    float* agg  = (float*)(ws + off); off = align_up(off + (size_t)nNodes * F_IN * sizeof(float), 256);
    float* Xt   = (float*)(ws + off); off = align_up(off + (size_t)nNodes * F_CAT * sizeof(float), 256);
    (void)ws_size;

    const int B = 256;

    // 1) degrees -> dinv
    hipMemsetAsync(dinv, 0, (size_t)nNodes * sizeof(float), stream);
    k_degree<<<(nEdges + B - 1) / B, B, 0, stream>>>(dst, dinv, nEdges);
    k_dsqrt<<<(nNodes + B - 1) / B, B, 0, stream>>>(dinv, nNodes);

    // 2) X0 = signal into Xt[:, 0:64]
    {
        int nt = nNodes * 16;
        k_copy_x0<<<(nt + B - 1) / B, B, 0, stream>>>(signal, Xt, nNodes);
    }

    // 3) Chebyshev recursion for X1..X3
    for (int t = 1; t < K_CHEB; ++t) {
        hipMemsetAsync(agg, 0, (size_t)nNodes * F_IN * sizeof(float), stream);
        {
            int nt = nEdges * 16;
            k_scatter<<<(nt + B - 1) / B, B, 0, stream>>>(
                src, dst, Xt + (t - 1) * F_IN, dinv, agg, nEdges);
        }
        {
            int nt = nNodes * 16;
            const float* Xprev = Xt + (t - 1) * F_IN;
            const float* Xpp   = (t >= 2) ? (Xt + (t - 2) * F_IN) : Xprev;  // gamma=0 on first step
            k_combine<<<(nt + B - 1) / B, B, 0, stream>>>(
                agg, dinv, lambda_max, Xprev, Xpp, Xt + t * F_IN, nNodes, t == 1 ? 1 : 0);
        }
    }

    // 4) GEMM + bias + ReLU via f32 WMMA (each wave: MT tiles sharing one B frag)
    {
        dim3 grid((nNodes + 16 * MT - 1) / (16 * MT), F_OUT / 128);
        k_gemm<<<grid, 256, 0, stream>>>(Xt, W, bias, out, nNodes);
    }
}